// CMLIF_44143673868829
// MI455X (gfx1250) — compile-verified
//
#include <hip/hip_runtime.h>

typedef __attribute__((ext_vector_type(2))) float v2f;
typedef __attribute__((ext_vector_type(8))) float v8f;
typedef __attribute__((ext_vector_type(4))) float f4;

#define TS 5
#define BS 32
#define CH 128
#define HH 64
#define WW 64
#define NPER (BS*CH*HH*WW)     // elements per time step = 16,777,216
#define N4   (NPER/4)          // float4 groups per time step

// ---------------------------------------------------------------------------
// Stage 1: P[c][j] = sum_{i=0..127} W[c][i][j]  (j = kh*3+kw, 9 positions)
// Done on the matrix engine: D = A x B with A[m][k] = W[c_block*16+m, i0+k, j]
// (fp32 16x4) and B = all-ones (4x16). Chained over 32 K=4 chunks via
// V_WMMA_F32_16X16X4_F32; every column of D carries P, so lanes 0 and 16
// write out the 16 channel sums. fp32 WMMA => exact-precision threshold.
// Two accumulators interleaved to cover the WMMA->WMMA RAW hazard latency.
// One full wave per block (EXEC must be all-ones for WMMA).
// ---------------------------------------------------------------------------
__global__ void __launch_bounds__(32)
psum_wmma_kernel(const float* __restrict__ w, float* __restrict__ P) {
    const int j    = blockIdx.x % 9;       // kernel position
    const int cb   = blockIdx.x / 9;       // channel block (0..7)
    const int lane = threadIdx.x;          // 0..31
    const int m    = lane & 15;
    const int kb   = (lane < 16) ? 0 : 2;  // A 16x4 f32 layout: lanes 16-31 hold K=2,3
    const int c    = cb * 16 + m;

    v8f acc0 = {};
    v8f acc1 = {};
    const v2f ones = {1.0f, 1.0f};

    #pragma unroll 4
    for (int i0 = 0; i0 < 128; i0 += 8) {
        v2f a0, a1;
        a0.x = w[(c * 128 + i0     + kb) * 9 + j];
        a0.y = w[(c * 128 + i0 + 1 + kb) * 9 + j];
        a1.x = w[(c * 128 + i0 + 4 + kb) * 9 + j];
        a1.y = w[(c * 128 + i0 + 5 + kb) * 9 + j];
        acc0 = __builtin_amdgcn_wmma_f32_16x16x4_f32(
                   false, a0, false, ones, (short)0, acc0, false, false);
        acc1 = __builtin_amdgcn_wmma_f32_16x16x4_f32(
                   false, a1, false, ones, (short)0, acc1, false, false);
    }

    // D layout (16x16 f32 C/D): VGPR r -> M=r (lanes 0-15), M=8+r (lanes 16-31);
    // lane index = N, all N columns identical here.
    if (lane == 0) {
        #pragma unroll
        for (int r = 0; r < 8; ++r)
            P[(cb * 16 + r) * 9 + j] = acc0[r] + acc1[r];
    } else if (lane == 16) {
        #pragma unroll
        for (int r = 0; r < 8; ++r)
            P[(cb * 16 + 8 + r) * 9 + j] = acc0[r] + acc1[r];
    }
}

// ---------------------------------------------------------------------------
// Stage 2: fold P + bias into the 9 border-case mask booleans per channel.
// sigmoid(z) > 0.5  <=>  z > 0. Case 0/2 in each axis drops one kernel row/col
// (SAME padding, cross-correlation orientation: h==0 drops kh=0, h==H-1 drops
// kh=2; same for w).
// ---------------------------------------------------------------------------
__global__ void mask_cases_kernel(const float* __restrict__ P,
                                  const float* __restrict__ bias,
                                  float* __restrict__ mask9) {
    int c = blockIdx.x * blockDim.x + threadIdx.x;
    if (c >= CH) return;
    float p[9];
    #pragma unroll
    for (int j = 0; j < 9; ++j) p[j] = P[c * 9 + j];
    float b = bias[c];
    #pragma unroll
    for (int hc = 0; hc < 3; ++hc) {
        int kh0 = (hc == 0) ? 1 : 0, kh1 = (hc == 2) ? 1 : 2;
        #pragma unroll
        for (int wc = 0; wc < 3; ++wc) {
            int kw0 = (wc == 0) ? 1 : 0, kw1 = (wc == 2) ? 1 : 2;
            float s = b;
            for (int kh = kh0; kh <= kh1; ++kh)
                for (int kw = kw0; kw <= kw1; ++kw)
                    s += p[kh * 3 + kw];
            mask9[c * 9 + hc * 3 + wc] = (s > 0.0f) ? 1.0f : 0.0f;
        }
    }
}

// ---------------------------------------------------------------------------
// Stage 3: streaming LIF. Bandwidth-bound (671 MB @ 23.3 TB/s ~ 29 us floor).
// One float4 (4 consecutive w) per thread; all 5 time-step loads issued up
// front for memory-level parallelism; non-temporal loads+stores keep the
// single-use 671 MB stream from thrashing the 192 MB L2.
// ---------------------------------------------------------------------------
__device__ __forceinline__ float lif_step(float& u, float xi) {
    // u' = TAU*u*(1 - (u>VTH)) + xi ; spike = (u' > VTH)
    float un = ((u > 1.0f) ? 0.0f : 0.5f) * u + xi;
    u = un;
    return (un > 1.0f) ? 1.0f : 0.0f;
}

__global__ void __launch_bounds__(256)
lif_kernel(const float* __restrict__ x, const float* __restrict__ mask9,
           float* __restrict__ out) {
    int idx4 = blockIdx.x * 256 + threadIdx.x;
    if (idx4 >= N4) return;
    int base = idx4 * 4;                 // element index within one time step
    int w0 = base & (WW - 1);            // 0,4,...,60
    int h  = (base >> 6) & (HH - 1);
    int c  = (base >> 12) & (CH - 1);

    int hc = (h == 0) ? 0 : ((h == HH - 1) ? 2 : 1);
    const float* mrow = mask9 + (c * 9 + hc * 3);
    float m_mid = mrow[1];
    float m_x = (w0 == 0)      ? mrow[0] : m_mid;
    float m_w = (w0 == WW - 4) ? mrow[2] : m_mid;

    const f4* x4 = (const f4*)x;
    f4*       o4 = (f4*)out;

    f4 xv[TS];
    #pragma unroll
    for (int t = 0; t < TS; ++t)
        xv[t] = __builtin_nontemporal_load(&x4[t * N4 + idx4]);

    // carried LIF state as plain scalars (vector elements can't bind to float&)
    float u0 = 0.0f, u1 = 0.0f, u2 = 0.0f, u3 = 0.0f;
    #pragma unroll
    for (int t = 0; t < TS; ++t) {
        f4 o;
        o.x = lif_step(u0, xv[t].x) * m_x;
        o.y = lif_step(u1, xv[t].y) * m_mid;
        o.z = lif_step(u2, xv[t].z) * m_mid;
        o.w = lif_step(u3, xv[t].w) * m_w;
        __builtin_nontemporal_store(o, &o4[t * N4 + idx4]);
    }
}

// ---------------------------------------------------------------------------
extern "C" void kernel_launch(void* const* d_in, const int* in_sizes, int n_in,
                              void* d_out, int out_size, void* d_ws, size_t ws_size,
                              hipStream_t stream) {
    const float* x    = (const float*)d_in[0];   // [T*bs, C, H, W]
    const float* conw = (const float*)d_in[1];   // [C, C, 3, 3]
    const float* conb = (const float*)d_in[2];   // [C]
    float* out = (float*)d_out;

    float* P     = (float*)d_ws;   // 128*9 floats
    float* mask9 = P + CH * 9;     // 128*9 floats

    psum_wmma_kernel<<<72, 32, 0, stream>>>(conw, P);
    mask_cases_kernel<<<1, 128, 0, stream>>>(P, conb, mask9);
    lif_kernel<<<N4 / 256, 256, 0, stream>>>(x, mask9, out);
}